// TopKRouter_43731357008648
// MI455X (gfx1250) — compile-verified
//
#include <hip/hip_runtime.h>
#include <hip/hip_bf16.h>

typedef float v2f __attribute__((ext_vector_type(2)));
typedef float v8f __attribute__((ext_vector_type(8)));
typedef int   v4i __attribute__((ext_vector_type(4)));

#define NT 16384      // tokens
#define DM 4096       // d_model
#define NE 64         // experts
#define NOISE_STD 0.1f

// CDNA5 async global->LDS copy builtins (ASYNCcnt path) — probe-confirmed present.
#if defined(__has_builtin)
#if __has_builtin(__builtin_amdgcn_global_load_async_to_lds_b128) && \
    __has_builtin(__builtin_amdgcn_s_wait_asynccnt)
#define HAVE_ASYNC_LDS 1
#endif
#endif
#ifndef HAVE_ASYNC_LDS
#define HAVE_ASYNC_LDS 0
#endif

#if HAVE_ASYNC_LDS
typedef __attribute__((address_space(1))) v4i* as1_v4i;
typedef __attribute__((address_space(3))) v4i* as3_v4i;
#endif

// ---------------- GEMM: logits = hidden @ rw^T + noise*0.1 ----------------
// 1024 blocks x 128 threads (4 waves). Block = 16 tokens, wave = 16 experts.
#define BK 64
#define LSTR 68   // LDS row stride (floats): conflict-free for wave32 float2 reads

__global__ __launch_bounds__(128) void moe_router_gemm(
    const float* __restrict__ hidden,   // [NT, DM]
    const float* __restrict__ rw,       // [NE, DM]
    const float* __restrict__ noise,    // [NT, NE]
    float* __restrict__ logits)         // [NT, NE]
{
    __shared__ __align__(16) float lA[16 * LSTR];
    __shared__ __align__(16) float lB[NE * LSTR];

    const int tid  = threadIdx.x;
    const int row0 = blockIdx.x * 16;
    const int wave = tid >> 5;
    const int lane = tid & 31;
    const int half = lane >> 4;     // which K-pair half
    const int lp   = lane & 15;     // M (for A) / N (for B) within tile
    const int e0   = wave * 16;     // expert tile base for this wave

    const int lr = tid >> 4;        // 0..7 cooperative-load row
    const int lc = (tid & 15) * 4;  // 0..60 cooperative-load col (float4)

    v8f acc = {};

    for (int k0 = 0; k0 < DM; k0 += BK) {
#if HAVE_ASYNC_LDS
        // Direct async global->LDS copies (no VGPR bounce), tracked by ASYNCcnt.
        #pragma unroll
        for (int i = 0; i < 2; ++i) {
            const int r = lr + 8 * i;
            __builtin_amdgcn_global_load_async_to_lds_b128(
                (as1_v4i)(hidden + (size_t)(row0 + r) * DM + k0 + lc),
                (as3_v4i)&lA[r * LSTR + lc], 0, 0);
        }
        #pragma unroll
        for (int i = 0; i < 8; ++i) {
            const int e = lr + 8 * i;
            __builtin_amdgcn_global_load_async_to_lds_b128(
                (as1_v4i)(rw + (size_t)e * DM + k0 + lc),
                (as3_v4i)&lB[e * LSTR + lc], 0, 0);
        }
        __builtin_amdgcn_s_wait_asynccnt(0);
#else
        // Stage A: 16 rows x 64 K-floats (2 float4 per thread)
        #pragma unroll
        for (int i = 0; i < 2; ++i) {
            const int r = lr + 8 * i;
            float4 v = *(const float4*)(hidden + (size_t)(row0 + r) * DM + k0 + lc);
            lA[r * LSTR + lc + 0] = v.x;
            lA[r * LSTR + lc + 1] = v.y;
            lA[r * LSTR + lc + 2] = v.z;
            lA[r * LSTR + lc + 3] = v.w;
        }
        // Stage B: 64 experts x 64 K-floats (8 float4 per thread)
        #pragma unroll
        for (int i = 0; i < 8; ++i) {
            const int e = lr + 8 * i;
            float4 v = *(const float4*)(rw + (size_t)e * DM + k0 + lc);
            lB[e * LSTR + lc + 0] = v.x;
            lB[e * LSTR + lc + 1] = v.y;
            lB[e * LSTR + lc + 2] = v.z;
            lB[e * LSTR + lc + 3] = v.w;
        }
        // Prefetch next A chunk while WMMAs run (global_prefetch_b8)
        if (k0 + BK < DM) {
            __builtin_prefetch(hidden + (size_t)(row0 + lr) * DM + k0 + BK + lc, 0, 1);
        }
#endif
        __syncthreads();

        // 16 x V_WMMA_F32_16X16X4_F32 over this K-chunk
        #pragma unroll
        for (int kk = 0; kk < BK; kk += 4) {
            v2f a = *(const v2f*)&lA[lp * LSTR + kk + 2 * half];
            v2f b = *(const v2f*)&lB[(e0 + lp) * LSTR + kk + 2 * half];
            acc = __builtin_amdgcn_wmma_f32_16x16x4_f32(
                false, a, false, b, (short)0, acc, false, false);
        }
        __syncthreads();
    }

    // Epilogue: C/D layout — VGPR j, lane(half,lp): M = j + 8*half, N = lp
    #pragma unroll
    for (int j = 0; j < 8; ++j) {
        const int m = j + 8 * half;
        const size_t g = (size_t)(row0 + m) * NE + e0 + lp;
        logits[g] = acc[j] + noise[g] * NOISE_STD;
    }
}

// ---------------- Top-2 + softmax + deterministic aux accumulation ----------------
// 256 blocks x 256 threads (8 waves); each wave handles 8 tokens sequentially.
__global__ __launch_bounds__(256) void moe_router_topk(
    const float* __restrict__ logits,
    float* __restrict__ idx_out,            // [NT,2] (indices stored as float)
    float* __restrict__ w_out,              // [NT,2]
    unsigned long long* __restrict__ probAcc, // [NE] fixed-point 32.32
    unsigned int* __restrict__ cntAcc)        // [NE]
{
    __shared__ unsigned long long sProb[NE];
    __shared__ unsigned int sCnt[NE];

    const int tid  = threadIdx.x;
    const int lane = tid & 31;
    const int wv   = tid >> 5;

    if (tid < NE) { sProb[tid] = 0ull; sCnt[tid] = 0u; }
    __syncthreads();

    for (int iter = 0; iter < 8; ++iter) {
        const int t = iter * 2048 + blockIdx.x * 8 + wv;   // 0..16383
        const float a0 = logits[(size_t)t * NE + lane];
        const float a1 = logits[(size_t)t * NE + 32 + lane];

        // local sorted top-2
        float v0 = a0, v1 = a1;
        int   i0 = lane, i1 = lane + 32;
        if (v1 > v0) { float tv = v0; v0 = v1; v1 = tv; int ti = i0; i0 = i1; i1 = ti; }

        // wave32 butterfly merge of sorted pairs
        #pragma unroll
        for (int off = 16; off > 0; off >>= 1) {
            float ov0 = __shfl_xor(v0, off, 32);
            int   oi0 = __shfl_xor(i0, off, 32);
            float ov1 = __shfl_xor(v1, off, 32);
            int   oi1 = __shfl_xor(i1, off, 32);
            if (ov0 > v0) {
                if (ov1 > v0) { v1 = ov1; i1 = oi1; } else { v1 = v0; i1 = i0; }
                v0 = ov0; i0 = oi0;
            } else if (ov0 > v1) {
                v1 = ov0; i1 = oi0;
            }
        }

        // full softmax over 64 logits (max == v0)
        const float p0 = __expf(a0 - v0);
        const float p1 = __expf(a1 - v0);
        float s = p0 + p1;
        #pragma unroll
        for (int off = 16; off > 0; off >>= 1) s += __shfl_xor(s, off, 32);
        const float inv = 1.0f / s;

        // deterministic fixed-point accumulation (integer atomics are associative)
        atomicAdd(&sProb[lane],      (unsigned long long)(p0 * inv * 4294967296.0f));
        atomicAdd(&sProb[lane + 32], (unsigned long long)(p1 * inv * 4294967296.0f));

        if (lane == 0) {
            atomicAdd(&sCnt[i0], 1u);
            atomicAdd(&sCnt[i1], 1u);
            const float e1 = __expf(v1 - v0);     // softmax over the 2 selected logits
            const float w0 = 1.0f / (1.0f + e1);
            idx_out[2 * t]     = (float)i0;
            idx_out[2 * t + 1] = (float)i1;
            w_out[2 * t]       = w0;
            w_out[2 * t + 1]   = e1 * w0;
        }
    }
    __syncthreads();
    if (tid < NE) {
        atomicAdd(&probAcc[tid], sProb[tid]);
        atomicAdd(&cntAcc[tid], sCnt[tid]);
    }
}

// ---------------- Final aux-loss reduction ----------------
__global__ void moe_aux(const unsigned long long* __restrict__ probAcc,
                        const unsigned int* __restrict__ cntAcc,
                        float* __restrict__ aux)
{
    const int lane = threadIdx.x;  // 32 threads
    float v = 0.0f;
    #pragma unroll
    for (int e = lane; e < NE; e += 32) {
        const float meanp = (float)((double)probAcc[e] * (1.0 / 4294967296.0)) / (float)NT;
        const float usage = (float)cntAcc[e] / ((float)NT * 2.0f);
        v += usage * meanp;
    }
    #pragma unroll
    for (int off = 16; off > 0; off >>= 1) v += __shfl_xor(v, off, 32);
    if (lane == 0) aux[0] = v * (float)NE * 1e-3f;
}

__global__ void zero_ws(unsigned int* p) {
    if (threadIdx.x < 192) p[threadIdx.x] = 0u;   // 64 u64 + 64 u32 = 768 B
}

extern "C" void kernel_launch(void* const* d_in, const int* in_sizes, int n_in,
                              void* d_out, int out_size, void* d_ws, size_t ws_size,
                              hipStream_t stream) {
    const float* hidden = (const float*)d_in[0];   // [16384,4096]
    const float* rw     = (const float*)d_in[1];   // [64,4096]
    const float* noise  = (const float*)d_in[2];   // [16384,64]

    float* out     = (float*)d_out;
    float* idx_out = out;                       // 32768
    float* w_out   = out + NT * 2;              // 32768
    float* logits  = out + NT * 4;              // 1048576
    float* aux     = out + NT * 4 + NT * NE;    // 1

    unsigned long long* probAcc = (unsigned long long*)d_ws;
    unsigned int*       cntAcc  = (unsigned int*)((char*)d_ws + NE * sizeof(unsigned long long));

    zero_ws<<<1, 256, 0, stream>>>((unsigned int*)d_ws);
    moe_router_gemm<<<NT / 16, 128, 0, stream>>>(hidden, rw, noise, logits);
    moe_router_topk<<<256, 256, 0, stream>>>(logits, idx_out, w_out, probAcc, cntAcc);
    moe_aux<<<1, 32, 0, stream>>>(probAcc, cntAcc, aux);
}